// AbsoluteNeuralLayer_50964081935036
// MI455X (gfx1250) — compile-verified
//
#include <hip/hip_runtime.h>
#include <math.h>

typedef float v2f __attribute__((ext_vector_type(2)));
typedef float v8f __attribute__((ext_vector_type(8)));

static constexpr int BATCH = 4096;
static constexpr int DIN   = 2048;
static constexpr int DOUTC = 2048;
static constexpr int DEPTH = 16;

static constexpr int BM = 128;
static constexpr int BN = 128;
static constexpr int BK = 32;
static constexpr int LDA = 36;   // padded stride for As (multiple of 4, conflict-free)
static constexpr int LDB = 144;  // padded stride for Bs (multiple of 16, conflict-free)
static constexpr int NT  = DIN / BK;  // 64 K-tiles

// ---- CDNA5 async global->LDS copy (ASYNCcnt-tracked, no VGPR staging) ----
__device__ __forceinline__ void async_copy_b128(unsigned lds_off, const float* g) {
    asm volatile("global_load_async_to_lds_b128 %0, %1, off"
                 :: "v"(lds_off), "v"((unsigned long long)(uintptr_t)g)
                 : "memory");
}
__device__ __forceinline__ void wait_async_le8() {
    asm volatile("s_wait_asynccnt 0x8" ::: "memory");
}
__device__ __forceinline__ void wait_async_0() {
    asm volatile("s_wait_asynccnt 0x0" ::: "memory");
}

// probs[j] = ( prod_{d,g} cos(W[d, j, g]) )^2 / DIN — exact collapse of the scan:
// roll() of a uniform vector is identity, so the sin terms cancel each step.
__global__ __launch_bounds__(256)
void absprobs_kernel(const float* __restrict__ W, float* __restrict__ probs) {
    const int j = blockIdx.x * 256 + threadIdx.x;
    if (j >= DOUTC) return;
    float p = 1.0f / sqrtf((float)DIN);
    #pragma unroll
    for (int d = 0; d < DEPTH; ++d) {
        const float* w = W + ((size_t)d * DIN + j) * DOUTC;  // W[d, j, 0..2]
        p *= cosf(w[0]);
        p *= cosf(w[1]);
        p *= cosf(w[2]);
    }
    probs[j] = p * p;
}

__global__ __launch_bounds__(256)
void gemm_tanh_kernel(const float* __restrict__ X,
                      const float* __restrict__ Wc,
                      const float* __restrict__ bias,
                      const float* __restrict__ probs,
                      float* __restrict__ out) {
    __shared__ float Abuf[3][BM * LDA];   // 3 x 18 KB
    __shared__ float Bbuf[3][BK * LDB];   // 3 x 18 KB

    const int tid   = threadIdx.x;
    const int lane  = tid & 31;
    const int wave  = tid >> 5;
    const int l16   = lane & 15;
    const int khalf = (lane >> 4) << 1;  // 0 or 2: K offset for A/B fragments
    const int mhalf = (lane >> 4) << 3;  // 0 or 8: M offset for C/D rows

    const int bm = blockIdx.y * BM;
    const int bn = blockIdx.x * BN;
    const int wm = (wave >> 2) * 64;     // 2 waves along M
    const int wn = (wave & 3) * 32;      // 4 waves along N

    // global->LDS staging assignments (16 floats per thread per matrix)
    const int arow = tid >> 1;           // 0..127
    const int acol = (tid & 1) << 4;     // 0 or 16
    const int brow = tid >> 3;           // 0..31
    const int bcol = (tid & 7) << 4;     // 0..112

    const float* agp = X  + (size_t)(bm + arow) * DIN + acol;
    const float* bgp = Wc + (size_t)brow * DOUTC + bn + bcol;

    // LDS byte offsets: generic __shared__ pointer = {aperture_hi, lds_offset}
    const unsigned aoff = (unsigned)(uintptr_t)&Abuf[0][arow * LDA + acol];
    const unsigned boff = (unsigned)(uintptr_t)&Bbuf[0][brow * LDB + bcol];
    constexpr unsigned ASTRIDE = (unsigned)(BM * LDA * sizeof(float));
    constexpr unsigned BSTRIDE = (unsigned)(BK * LDB * sizeof(float));

    auto issue_tile = [&](int t) {
        const unsigned buf = (unsigned)(t % 3);
        const float* ga = agp + t * BK;
        const float* gb = bgp + (size_t)(t * BK) * DOUTC;
        #pragma unroll
        for (int i = 0; i < 4; ++i)
            async_copy_b128(aoff + buf * ASTRIDE + (unsigned)(i * 16), ga + i * 4);
        #pragma unroll
        for (int i = 0; i < 4; ++i)
            async_copy_b128(boff + buf * BSTRIDE + (unsigned)(i * 16), gb + i * 4);
    };

    const v8f vzero = {0.f, 0.f, 0.f, 0.f, 0.f, 0.f, 0.f, 0.f};
    v8f acc[4][2];
    #pragma unroll
    for (int mi = 0; mi < 4; ++mi)
        #pragma unroll
        for (int ni = 0; ni < 2; ++ni)
            acc[mi][ni] = vzero;

    // Prefetch two tiles ahead (8 async ops each, in-order completion).
    issue_tile(0);
    issue_tile(1);

    for (int t = 0; t < NT; ++t) {
        // Per-wave: tile t's 8 copies complete when ASYNCcnt <= 8 (or 0 at tail).
        if (t + 1 < NT) wait_async_le8(); else wait_async_0();
        // Barrier: everyone's tile-t data is in LDS AND everyone finished
        // compute(t-1), so buf[(t+2)%3] == buf[(t-1)%3] is safe to overwrite.
        __syncthreads();
        if (t + 2 < NT) issue_tile(t + 2);

        const float* As = Abuf[t % 3];
        const float* Bs = Bbuf[t % 3];

        #pragma unroll
        for (int kk = 0; kk < BK; kk += 4) {
            // A fragments: lane holds (M = l16, K = khalf + v) — one b64 LDS load
            v2f a[4];
            #pragma unroll
            for (int mi = 0; mi < 4; ++mi)
                a[mi] = *(const v2f*)&As[(wm + mi * 16 + l16) * LDA + kk + khalf];

            // B fragments: VGPR v holds (K = khalf + v, N = l16)
            v2f b[2];
            #pragma unroll
            for (int ni = 0; ni < 2; ++ni) {
                const int nb = wn + ni * 16 + l16;
                v2f tmp;
                tmp.x = Bs[(kk + khalf) * LDB + nb];
                tmp.y = Bs[(kk + khalf + 1) * LDB + nb];
                b[ni] = tmp;
            }

            #pragma unroll
            for (int mi = 0; mi < 4; ++mi)
                #pragma unroll
                for (int ni = 0; ni < 2; ++ni)
                    acc[mi][ni] = __builtin_amdgcn_wmma_f32_16x16x4_f32(
                        false, a[mi], false, b[ni],
                        (short)0, acc[mi][ni], false, false);
        }
    }

    // Epilogue: tanh(acc + bias[n] + probs[n]); C VGPR v = row (mhalf + v), col l16
    #pragma unroll
    for (int ni = 0; ni < 2; ++ni) {
        const int gn = bn + wn + ni * 16 + l16;
        const float add = bias[gn] + probs[gn];
        #pragma unroll
        for (int mi = 0; mi < 4; ++mi) {
            const int gmBase = bm + wm + mi * 16 + mhalf;
            #pragma unroll
            for (int v = 0; v < 8; ++v)
                out[(size_t)(gmBase + v) * DOUTC + gn] = tanhf(acc[mi][ni][v] + add);
        }
    }
}

extern "C" void kernel_launch(void* const* d_in, const int* in_sizes, int n_in,
                              void* d_out, int out_size, void* d_ws, size_t ws_size,
                              hipStream_t stream) {
    const float* x   = (const float*)d_in[0];  // [4096, 2048]
    const float* aw  = (const float*)d_in[1];  // [16, 2048, 2048]
    const float* cw  = (const float*)d_in[2];  // [2048, 2048]
    const float* cb  = (const float*)d_in[3];  // [2048]
    float* out   = (float*)d_out;              // [4096, 2048]
    float* probs = (float*)d_ws;               // 2048 floats of scratch

    absprobs_kernel<<<DOUTC / 256, 256, 0, stream>>>(aw, probs);

    dim3 grid(DOUTC / BN, BATCH / BM);  // 16 x 32 blocks
    gemm_tanh_kernel<<<grid, 256, 0, stream>>>(x, cw, cb, probs, out);
}